// HierarchicalSoftmaxTree_15865609191786
// MI455X (gfx1250) — compile-verified
//
#include <hip/hip_runtime.h>

// Hierarchical softmax:
//   out[b,t] = prod_{l<15} sigmoid( W[paths[ids[b,t],l]] . x[b] + bias[node] )
//
// Strategy (MI455X roofline): instead of gathering 503 MB of W rows, compute
// sigmoid logits for ALL 32767 internal nodes with one dense WMMA GEMM
// ([32767 x 1024] x [1024 x 8], W read once = 128 MB ~ 5.5us @ 23.3 TB/s),
// then do the cheap 1 MB path-gather + product out of L2.

#define HS_V   32768   // vocab (leaves); internal nodes = HS_V - 1
#define HS_D   15      // path depth
#define HS_DM  1024    // feature dim
#define HS_B   8       // batch
#define HS_T   1024    // targets per batch item

typedef float v2f __attribute__((ext_vector_type(2)));
typedef float v8f __attribute__((ext_vector_type(8)));

// ---------------------------------------------------------------------------
// Kernel 1: Z[node, b] = sigmoid(W[node] . x[b] + bias[node])  for all nodes.
// One wave32 per 16-node tile; V_WMMA_F32_16X16X4_F32, K-loop of 256 steps.
//   A tile (16x4 of W):  lane<16 -> row=lane, K0/K1 ; lane>=16 -> row=lane-16, K2/K3
//   B tile (4x16 of xT): lane<16 -> col=lane, K0/K1 ; lane>=16 -> col=lane-16, K2/K3
//   C tile (16x16):      VGPR j, lane<16 -> M=j, lane>=16 -> M=j+8 ; N=lane&15
// x is staged in LDS with one extra all-zero row backing the padded N=8..15.
// ---------------------------------------------------------------------------
__global__ __launch_bounds__(256, 2) void hs_node_sigmoid_kernel(
    const float* __restrict__ x,      // [HS_B, HS_DM]
    const float* __restrict__ W,      // [HS_V-1, HS_DM]
    const float* __restrict__ bias,   // [HS_V-1]
    float* __restrict__ Z)            // [HS_V, HS_B]  (row HS_V-1 is padding)
{
    __shared__ float xs[(HS_B + 1) * HS_DM];   // 36 KB: 8 real rows + 1 zero row

    const int tid = threadIdx.x;
    for (int i = tid; i < HS_B * HS_DM; i += 256) xs[i] = x[i];
    for (int i = tid; i < HS_DM;        i += 256) xs[HS_B * HS_DM + i] = 0.0f;
    __syncthreads();

    const int wave = tid >> 5;                 // 0..7
    const int lane = tid & 31;
    const int tile = blockIdx.x * 8 + wave;    // 0..2047
    const int row0 = tile * 16;
    const int m    = lane & 15;                // A: row-in-tile / B,C: column N
    const int hi   = lane >> 4;                // K sub-offset selector (0 or 1)

    // A operand base: W row (clamped: row 32767 doesn't exist, result unused)
    int rowA = row0 + m;
    if (rowA > HS_V - 2) rowA = HS_V - 2;
    const float* __restrict__ aptr = W + (size_t)rowA * HS_DM + hi * 2;

    // B operand base in LDS: column N = m (zero row for N >= 8)
    const int nrow = (m < HS_B) ? m : HS_B;
    const float* bptr = xs + nrow * HS_DM + hi * 2;

    v8f c = {};
    #pragma unroll 8
    for (int k = 0; k < HS_DM; k += 4) {
        v2f a = *(const v2f*)(aptr + k);       // global_load_b64, imm offset
        v2f b = *(const v2f*)(bptr + k);       // ds_load_b64, imm offset
        // D = A(16x4, f32) x B(4x16, f32) + C
        c = __builtin_amdgcn_wmma_f32_16x16x4_f32(
                /*neg_a=*/false, a, /*neg_b=*/false, b,
                /*c_mod=*/(short)0, c, /*reuse_a=*/false, /*reuse_b=*/false);
    }

    // Epilogue: bias + sigmoid, store valid columns (N < 8) of the C tile.
    const int rowBase = row0 + hi * 8;
    #pragma unroll
    for (int j = 0; j < 8; ++j) {
        int r  = rowBase + j;                       // node id (may be padded 32767)
        int rb = (r > HS_V - 2) ? (HS_V - 2) : r;   // clamp bias gather
        float logit = c[j] + bias[rb];
        float p = 1.0f / (1.0f + __expf(-logit));
        if (m < HS_B) Z[(size_t)r * HS_B + m] = p;
    }
}

// ---------------------------------------------------------------------------
// Kernel 2: out[b,t] = prod_{l<15} Z[paths[ids[b,t], l], b]
// 8192 outputs, 15 gathers each from the 1 MB L2-resident Z table.
// ---------------------------------------------------------------------------
__global__ __launch_bounds__(256) void hs_path_prod_kernel(
    const int*   __restrict__ ids,    // [HS_B, HS_T]
    const int*   __restrict__ paths,  // [HS_V, HS_D]
    const float* __restrict__ Z,      // [HS_V, HS_B]
    float* __restrict__ out)          // [HS_B, HS_T]
{
    const int i = blockIdx.x * 256 + threadIdx.x;
    if (i >= HS_B * HS_T) return;
    const int bcol = i >> 10;                  // i / HS_T
    const int id   = ids[i];
    const int* __restrict__ p = paths + (size_t)id * HS_D;
    float acc = 1.0f;
    #pragma unroll
    for (int l = 0; l < HS_D; ++l) {
        int node = p[l];
        acc *= Z[(size_t)node * HS_B + bcol];
    }
    out[i] = acc;
}

// ---------------------------------------------------------------------------
extern "C" void kernel_launch(void* const* d_in, const int* in_sizes, int n_in,
                              void* d_out, int out_size, void* d_ws, size_t ws_size,
                              hipStream_t stream) {
    const float* x     = (const float*)d_in[0];   // [8, 1024] f32
    const int*   ids   = (const int*)  d_in[1];   // [8, 1024] i32
    const int*   paths = (const int*)  d_in[2];   // [32768, 15] i32
    const float* W     = (const float*)d_in[3];   // [32767, 1024] f32
    const float* bias  = (const float*)d_in[4];   // [32767] f32
    float* out = (float*)d_out;                   // [8, 1024] f32
    float* Z   = (float*)d_ws;                    // [32768, 8] f32 = 1 MB scratch

    // Kernel 1: 2048 tiles of 16 nodes, 8 waves (tiles) per 256-thread block.
    hs_node_sigmoid_kernel<<<HS_V / 16 / 8, 256, 0, stream>>>(x, W, bias, Z);

    // Kernel 2: one thread per (b, t) output.
    hs_path_prod_kernel<<<(HS_B * HS_T) / 256, 256, 0, stream>>>(ids, paths, Z, out);
}